// ConditionedBilinearKoopmanTransition_30305289240688
// MI455X (gfx1250) — compile-verified
//
#include <hip/hip_runtime.h>

// ConditionedBilinearKoopmanTransition for MI455X (gfx1250, wave32).
// fp32 throughout via V_WMMA_F32_16X16X4_F32 (full-precision matrix pipe).
//
// b=4096, d=1024, u=32, r=32 (u*r = 1024), n_obs=25.
//
// Pipeline (all on `stream`, graph-capture safe):
//   k_linear    : d_out.z  = z @ A^T + (ut*dt) @ B^T
//   k_bilin_t   : ws.s     = (z @ NQmat) * ut_dt broadcast   (NQmat[j, m*32+r] = N_Q[m,j,r])
//   k_bilin_out : d_out.z += s @ NPmat                        (NPmat[m*32+r, i] = N_P[m,i,r])
//   k_obs       : d_out.y  = z_next @ C^T + ut_dt @ D^T       (tiny, VALU)
//
// Tiling: block = 8 waves as 4(M) x 2(N) -> 128x128 block tile.
// Wave tile = 32(M) x 64(N): per K-step of 4 -> 2 A-loads + 4 B-loads feed
// 8 WMMAs (0.75 vmem/wmma, vs 1.25 with a 16x64 wave tile).

typedef __attribute__((ext_vector_type(2))) float v2f;
typedef __attribute__((ext_vector_type(8))) float v8f;

#define BATCH 4096
#define DDIM  1024
#define UDIM  32
#define RDIM  32
#define NOBS  25

// WMMA fragment mapping (ISA 7.12.2, wave32):
//  A (16x4 f32):  lane L -> M = L&15 ; holds K = 2*(L>>4) + {0,1} in {v0,v1}
//  B (4x16 f32):  lane L -> N = L&15 ; holds K = 2*(L>>4) + {0,1} in {v0,v1}
//  C/D (16x16):   lane L, vgpr v -> M = v + 8*(L>>4), N = L&15

#define WMMA_F32(a, b, c) \
  __builtin_amdgcn_wmma_f32_16x16x4_f32(false, (a), false, (b), (short)0, (c), false, false)

__device__ __forceinline__ v2f ld2(const float* p) { return *(const v2f*)p; }

// ---------------------------------------------------------------------------
// z_linear = z @ A_bilinear^T + (ut*dt) @ B_bilinear^T  -> out [4096,1024]
// ---------------------------------------------------------------------------
__global__ __launch_bounds__(256) void k_linear(const float* __restrict__ z,
                                                const float* __restrict__ Ab,
                                                const float* __restrict__ Bb,
                                                const float* __restrict__ ut,
                                                const float* __restrict__ dtp,
                                                float* __restrict__ out) {
  const float dt = dtp[0];
  const int lane = threadIdx.x & 31;
  const int wave = threadIdx.x >> 5;
  const int wm = wave & 3;
  const int wn = wave >> 2;
  const int m0 = blockIdx.y * 128 + wm * 32;
  const int n0 = blockIdx.x * 128 + wn * 64;
  const int row = lane & 15;
  const int grp = lane >> 4;

  v8f acc[2][4] = {};

  // K = 1024 over z / A^T  (B[k,n] = Ab[n*1024 + k], contiguous in k)
  for (int k0 = 0; k0 < DDIM; k0 += 4) {
    const int kk = k0 + 2 * grp;
    v2f a0 = ld2(z + (m0 + row) * DDIM + kk);
    v2f a1 = ld2(z + (m0 + 16 + row) * DDIM + kk);
#pragma unroll
    for (int t = 0; t < 4; ++t) {
      const int n = n0 + 16 * t + row;
      v2f b = ld2(Ab + n * DDIM + kk);
      acc[0][t] = WMMA_F32(a0, b, acc[0][t]);
      acc[1][t] = WMMA_F32(a1, b, acc[1][t]);
    }
  }

  // K = 32 over ut*dt / B_bilinear^T  (B[k,n] = Bb[n*32 + k])
  for (int k0 = 0; k0 < UDIM; k0 += 4) {
    const int kk = k0 + 2 * grp;
    v2f a0 = ld2(ut + (m0 + row) * UDIM + kk);
    v2f a1 = ld2(ut + (m0 + 16 + row) * UDIM + kk);
    a0.x *= dt; a0.y *= dt;
    a1.x *= dt; a1.y *= dt;
#pragma unroll
    for (int t = 0; t < 4; ++t) {
      const int n = n0 + 16 * t + row;
      v2f b = ld2(Bb + n * UDIM + kk);
      acc[0][t] = WMMA_F32(a0, b, acc[0][t]);
      acc[1][t] = WMMA_F32(a1, b, acc[1][t]);
    }
  }

#pragma unroll
  for (int mt = 0; mt < 2; ++mt)
#pragma unroll
    for (int t = 0; t < 4; ++t)
#pragma unroll
      for (int v = 0; v < 8; ++v)
        out[(m0 + 16 * mt + 8 * grp + v) * DDIM + n0 + 16 * t + row] = acc[mt][t][v];
}

// ---------------------------------------------------------------------------
// s = (z @ NQmat) * ut_dt broadcast over r.
// NQmat[k, n] = N_Q[n>>5, k, n&31]  (element at (n>>5)*32768 + k*32 + (n&31))
// ---------------------------------------------------------------------------
__global__ __launch_bounds__(256) void k_bilin_t(const float* __restrict__ z,
                                                 const float* __restrict__ NQ,
                                                 const float* __restrict__ ut,
                                                 const float* __restrict__ dtp,
                                                 float* __restrict__ s) {
  const int lane = threadIdx.x & 31;
  const int wave = threadIdx.x >> 5;
  const int wm = wave & 3;
  const int wn = wave >> 2;
  const int m0 = blockIdx.y * 128 + wm * 32;
  const int n0 = blockIdx.x * 128 + wn * 64;
  const int row = lane & 15;
  const int grp = lane >> 4;

  v8f acc[2][4] = {};

  for (int k0 = 0; k0 < DDIM; k0 += 4) {
    const int kk = k0 + 2 * grp;
    v2f a0 = ld2(z + (m0 + row) * DDIM + kk);
    v2f a1 = ld2(z + (m0 + 16 + row) * DDIM + kk);
#pragma unroll
    for (int t = 0; t < 4; ++t) {
      const int n = n0 + 16 * t + row;
      const float* bp = NQ + (n >> 5) * (DDIM * RDIM) + kk * RDIM + (n & 31);
      v2f b;
      b.x = bp[0];       // k = kk
      b.y = bp[RDIM];    // k = kk+1 (stride 32 floats)
      acc[0][t] = WMMA_F32(a0, b, acc[0][t]);
      acc[1][t] = WMMA_F32(a1, b, acc[1][t]);
    }
  }

  const float dt = dtp[0];
#pragma unroll
  for (int mt = 0; mt < 2; ++mt)
#pragma unroll
    for (int t = 0; t < 4; ++t) {
      const int n = n0 + 16 * t + row;
      const int mi = n >> 5;
#pragma unroll
      for (int v = 0; v < 8; ++v) {
        const int br = m0 + 16 * mt + 8 * grp + v;  // batch row of this element
        s[br * DDIM + n] = acc[mt][t][v] * ut[br * UDIM + mi] * dt;
      }
    }
}

// ---------------------------------------------------------------------------
// out += s @ NPmat.
// NPmat[k, i] = N_P[k>>5, i, k&31]  (element at (k>>5)*32768 + i*32 + (k&31));
// kk = k0+2*grp with k0%4==0 => (kk&31) even and <=30, so {kk, kk+1} are
// contiguous floats inside one m-block -> single 8-byte load.
// ---------------------------------------------------------------------------
__global__ __launch_bounds__(256) void k_bilin_out(const float* __restrict__ s,
                                                   const float* __restrict__ NP,
                                                   float* __restrict__ out) {
  const int lane = threadIdx.x & 31;
  const int wave = threadIdx.x >> 5;
  const int wm = wave & 3;
  const int wn = wave >> 2;
  const int m0 = blockIdx.y * 128 + wm * 32;
  const int n0 = blockIdx.x * 128 + wn * 64;
  const int row = lane & 15;
  const int grp = lane >> 4;

  v8f acc[2][4] = {};

  for (int k0 = 0; k0 < DDIM; k0 += 4) {
    const int kk = k0 + 2 * grp;
    v2f a0 = ld2(s + (m0 + row) * DDIM + kk);
    v2f a1 = ld2(s + (m0 + 16 + row) * DDIM + kk);
#pragma unroll
    for (int t = 0; t < 4; ++t) {
      const int n = n0 + 16 * t + row;
      v2f b = ld2(NP + (kk >> 5) * (DDIM * RDIM) + n * RDIM + (kk & 31));
      acc[0][t] = WMMA_F32(a0, b, acc[0][t]);
      acc[1][t] = WMMA_F32(a1, b, acc[1][t]);
    }
  }

#pragma unroll
  for (int mt = 0; mt < 2; ++mt)
#pragma unroll
    for (int t = 0; t < 4; ++t)
#pragma unroll
      for (int v = 0; v < 8; ++v) {
        const int idx = (m0 + 16 * mt + 8 * grp + v) * DDIM + n0 + 16 * t + row;
        out[idx] += acc[mt][t][v];  // z_linear already there (k_linear ran earlier)
      }
}

// ---------------------------------------------------------------------------
// yt = z_next @ C^T + (ut*dt) @ D^T   [4096, 25] — negligible FLOPs, VALU.
// One wave per batch row; lane o < 25 computes yt[b, o].
// ---------------------------------------------------------------------------
__global__ __launch_bounds__(32) void k_obs(const float* __restrict__ znext,
                                            const float* __restrict__ ut,
                                            const float* __restrict__ dtp,
                                            const float* __restrict__ Cm,
                                            const float* __restrict__ Dm,
                                            float* __restrict__ yt) {
  const int b = blockIdx.x;
  const int o = threadIdx.x;
  if (o >= NOBS) return;
  const float dt = dtp[0];
  const float* zr = znext + (size_t)b * DDIM;
  const float* cr = Cm + o * DDIM;
  float acc = 0.0f;
  for (int k = 0; k < DDIM; ++k) acc = fmaf(zr[k], cr[k], acc);
  const float* ur = ut + b * UDIM;
  const float* dr = Dm + o * UDIM;
  float acc2 = 0.0f;
  for (int j = 0; j < UDIM; ++j) acc2 = fmaf(ur[j], dr[j], acc2);
  yt[b * NOBS + o] = acc + acc2 * dt;
}

// ---------------------------------------------------------------------------
extern "C" void kernel_launch(void* const* d_in, const int* in_sizes, int n_in,
                              void* d_out, int out_size, void* d_ws, size_t ws_size,
                              hipStream_t stream) {
  const float* z  = (const float*)d_in[0];
  // d_in[1] = z_static (unused by the reference math)
  const float* dt = (const float*)d_in[2];
  const float* ut = (const float*)d_in[3];
  const float* Ab = (const float*)d_in[4];
  const float* Bb = (const float*)d_in[5];
  const float* NP = (const float*)d_in[6];
  const float* NQ = (const float*)d_in[7];
  const float* Cm = (const float*)d_in[8];
  const float* Dm = (const float*)d_in[9];

  float* z_next = (float*)d_out;                            // [4096, 1024]
  float* yt     = (float*)d_out + (size_t)BATCH * DDIM;     // [4096, 25]
  float* s      = (float*)d_ws;                             // [4096, 1024] scratch (16 MB)

  dim3 grid(DDIM / 128, BATCH / 128);  // (8, 32)
  dim3 blk(256);                       // 8 waves: 4(M) x 2(N)

  k_linear<<<grid, blk, 0, stream>>>(z, Ab, Bb, ut, dt, z_next);
  k_bilin_t<<<grid, blk, 0, stream>>>(z, NQ, ut, dt, s);
  k_bilin_out<<<grid, blk, 0, stream>>>(s, NP, z_next);
  k_obs<<<dim3(BATCH), dim3(32), 0, stream>>>(z_next, ut, dt, Cm, Dm, yt);
}